// VascularGAT_34127810134072
// MI455X (gfx1250) — compile-verified
//
#include <hip/hip_runtime.h>
#include <hip/hip_bf16.h>
#include <math.h>

// ---------------------------------------------------------------------------
// CDNA5 (gfx1250) GAT forward. fp32 path; GEMMs on v_wmma_f32_16x16x4_f32.
// ---------------------------------------------------------------------------

typedef __attribute__((ext_vector_type(2))) float v2f;
typedef __attribute__((ext_vector_type(8))) float v8f;

#define HID 64

// ---------------- utility ----------------
__global__ void fill_f32(float* __restrict__ p, float v, int count) {
    int i = blockIdx.x * blockDim.x + threadIdx.x;
    if (i < count) p[i] = v;
}

__device__ __forceinline__ void atomicMaxF(float* addr, float value) {
    // Order-preserving int encoding trick; works for mixed signs with -inf init.
    if (value >= 0.0f) atomicMax((int*)addr, __float_as_int(value));
    else               atomicMin((unsigned int*)addr, __float_as_uint(value));
}

// ---------------- GEMM: H[n,64] = X[n,K] @ W[K,64], via WMMA f32 16x16x4 ----
// block = 128 threads = 4 waves; wave w computes columns [16w,16w+16) of a
// 16-row tile. A tile staged in LDS (padded to kill bank conflicts).
__global__ void gemm_wmma(const float* __restrict__ X, const float* __restrict__ W,
                          float* __restrict__ H, int n, int K) {
    __shared__ float As[16][68];
    const int row0 = blockIdx.x * 16;
    const int tid  = threadIdx.x;

    for (int i = tid; i < 16 * K; i += blockDim.x) {
        int r = i / K, c = i - r * K;
        int gr = row0 + r;
        As[r][c] = (gr < n) ? X[gr * K + c] : 0.0f;
    }
    __syncthreads();

    const int lane  = tid & 31;
    const int wave  = tid >> 5;
    const int col0  = wave * 16;
    const int lh    = lane & 15;          // M row (A) / N col (B,D)
    const int khi   = (lane >> 4) * 2;    // K sub-offset: 0 or 2

    v8f c = {};
    for (int k0 = 0; k0 < K; k0 += 4) {
        v2f a, b;
        a[0] = As[lh][k0 + khi];
        a[1] = As[lh][k0 + khi + 1];
        b[0] = W[(k0 + khi)     * HID + col0 + lh];
        b[1] = W[(k0 + khi + 1) * HID + col0 + lh];
        c = __builtin_amdgcn_wmma_f32_16x16x4_f32(false, a, false, b,
                                                  (short)0, c, false, false);
    }
    // D layout: VGPR r -> (M = r + 8*(lane>=16), N = lane&15)
    const int rbase = row0 + ((lane >> 4) << 3);
#pragma unroll
    for (int r = 0; r < 8; ++r) {
        int gr = rbase + r;
        if (gr < n) H[gr * HID + col0 + lh] = c[r];
    }
}

// ---------------- attention coefficients: als/ald[n,heads] -----------------
__global__ void attn_coef(const float* __restrict__ H,
                          const float* __restrict__ asrc, const float* __restrict__ adst,
                          float* __restrict__ als, float* __restrict__ ald,
                          int n, int heads, int C) {
    int i = blockIdx.x * blockDim.x + threadIdx.x;
    if (i >= n * heads) return;
    int node = i / heads, h = i - node * heads;
    const float* hp = H + node * HID + h * C;
    const float* ap = asrc + h * C;
    const float* bp = adst + h * C;
    float s = 0.0f, d = 0.0f;
    for (int c = 0; c < C; ++c) { float v = hp[c]; s += v * ap[c]; d += v * bp[c]; }
    als[i] = s;
    ald[i] = d;
}

// ---------------- edge pass 1: leaky-relu score + segment max --------------
__global__ void edge_scores(const int* __restrict__ ei,
                            const float* __restrict__ als, const float* __restrict__ ald,
                            float* __restrict__ m, float* __restrict__ eex,
                            int ET, int Eraw, int heads) {
    int idx = blockIdx.x * blockDim.x + threadIdx.x;
    if (idx >= ET * heads) return;
    int e = idx / heads, h = idx - e * heads;
    int s, d;
    if (e < Eraw) { s = ei[e]; d = ei[Eraw + e]; } else { s = d = e - Eraw; }
    float v = als[s * heads + h] + ald[d * heads + h];
    v = v > 0.0f ? v : 0.2f * v;               // leaky_relu(0.2)
    eex[idx] = v;
    atomicMaxF(&m[d * heads + h], v);
}

// ---------------- edge pass 2: exp + segment sum ---------------------------
__global__ void edge_expsum(const int* __restrict__ ei,
                            float* __restrict__ eex, const float* __restrict__ m,
                            float* __restrict__ ssum,
                            int ET, int Eraw, int heads) {
    int idx = blockIdx.x * blockDim.x + threadIdx.x;
    if (idx >= ET * heads) return;
    int e = idx / heads, h = idx - e * heads;
    int d;
    if (e < Eraw) { d = ei[Eraw + e]; } else { d = e - Eraw; }
    float ex = expf(eex[idx] - m[d * heads + h]);
    eex[idx] = ex;
    atomicAdd(&ssum[d * heads + h], ex);
}

// ---------------- edge pass 3: alpha-weighted gather/scatter ---------------
__global__ void edge_aggr(const int* __restrict__ ei, const float* __restrict__ Hfeat,
                          const float* __restrict__ eex, const float* __restrict__ ssum,
                          float* __restrict__ agg,
                          int ET, int Eraw, int heads, int C) {
    int idx = blockIdx.x * blockDim.x + threadIdx.x;
    if (idx >= ET * heads) return;
    int e = idx / heads, h = idx - e * heads;
    int s, d;
    if (e < Eraw) { s = ei[e]; d = ei[Eraw + e]; } else { s = d = e - Eraw; }
    float alpha = eex[idx] / (ssum[d * heads + h] + 1e-16f);
    const float4* hp4 = reinterpret_cast<const float4*>(Hfeat + s * HID + h * C);
    float* op = agg + d * HID + h * C;
    int q = C >> 2;
    for (int c4 = 0; c4 < q; ++c4) {
        float4 v = hp4[c4];
        atomicAdd(op + 4 * c4 + 0, v.x * alpha);
        atomicAdd(op + 4 * c4 + 1, v.y * alpha);
        atomicAdd(op + 4 * c4 + 2, v.z * alpha);
        atomicAdd(op + 4 * c4 + 3, v.w * alpha);
    }
}

// ---------------- BatchNorm stats (per-channel sum / sumsq) ----------------
__global__ void bn_stats(const float* __restrict__ X,
                         float* __restrict__ gsum, float* __restrict__ gsq, int n) {
    __shared__ float ssum[HID], ssq[HID];
    int tid = threadIdx.x;
    if (tid < HID) { ssum[tid] = 0.0f; ssq[tid] = 0.0f; }
    __syncthreads();
    int total = n * HID;
    for (int i = blockIdx.x * blockDim.x + tid; i < total; i += gridDim.x * blockDim.x) {
        float v = X[i];
        int ch = i & (HID - 1);
        atomicAdd(&ssum[ch], v);
        atomicAdd(&ssq[ch], v * v);
    }
    __syncthreads();
    if (tid < HID) {
        atomicAdd(&gsum[tid], ssum[tid]);
        atomicAdd(&gsq[tid],  ssq[tid]);
    }
}

// ---------------- BatchNorm normalize + ELU --------------------------------
// NOTE: GAT bias b is a per-channel constant added before BN; BN subtracts the
// per-channel mean, so b cancels exactly and is intentionally omitted.
__global__ void bn_elu(const float* __restrict__ X,
                       const float* __restrict__ gsum, const float* __restrict__ gsq,
                       const float* __restrict__ gamma, const float* __restrict__ beta,
                       float* __restrict__ Y, int n) {
    int i = blockIdx.x * blockDim.x + threadIdx.x;
    if (i >= n * HID) return;
    int ch = i & (HID - 1);
    float invn = 1.0f / (float)n;
    float mu  = gsum[ch] * invn;
    float var = gsq[ch] * invn - mu * mu;
    float v = (X[i] - mu) * rsqrtf(var + 1e-5f) * gamma[ch] + beta[ch];
    Y[i] = v > 0.0f ? v : expm1f(v);
}

// ---------------- global mean pool -----------------------------------------
__global__ void pool_kernel(const float* __restrict__ nodeEmb, const int* __restrict__ batch,
                            float* __restrict__ psum, float* __restrict__ pcnt, int n) {
    int i = blockIdx.x * blockDim.x + threadIdx.x;
    if (i >= n * HID) return;
    int node = i >> 6, ch = i & (HID - 1);
    int g = batch[node];
    atomicAdd(&psum[g * HID + ch], nodeEmb[i]);
    if (ch == 0) atomicAdd(&pcnt[g], 1.0f);
}

// ---------------- classifier + embedding heads -----------------------------
__global__ void head_kernel(const float* __restrict__ psum, const float* __restrict__ pcnt,
                            const float* __restrict__ Wc1, const float* __restrict__ bc1,
                            const float* __restrict__ Wc2, const float* __restrict__ bc2,
                            const float* __restrict__ We,  const float* __restrict__ bee,
                            float* __restrict__ out_logits, float* __restrict__ out_emb,
                            int Gn) {
    int g = blockIdx.x * blockDim.x + threadIdx.x;
    if (g >= Gn) return;
    float ge[HID];
    float cnt = fmaxf(pcnt[g], 1.0f);
    for (int c = 0; c < HID; ++c) ge[c] = psum[g * HID + c] / cnt;
    float hid[32];
    for (int j = 0; j < 32; ++j) {
        float acc = bc1[j];
        for (int c = 0; c < HID; ++c) acc += ge[c] * Wc1[c * 32 + j];
        hid[j] = fmaxf(acc, 0.0f);
    }
    for (int j = 0; j < 2; ++j) {
        float acc = bc2[j];
        for (int c = 0; c < 32; ++c) acc += hid[c] * Wc2[c * 2 + j];
        out_logits[g * 2 + j] = acc;
    }
    for (int j = 0; j < HID; ++j) {
        float acc = bee[j];
        for (int c = 0; c < HID; ++c) acc += ge[c] * We[c * HID + j];
        out_emb[g * HID + j] = acc;
    }
}

// ---------------------------------------------------------------------------
extern "C" void kernel_launch(void* const* d_in, const int* in_sizes, int n_in,
                              void* d_out, int out_size, void* d_ws, size_t ws_size,
                              hipStream_t stream) {
    // inputs (setup_inputs dict order)
    const float* x    = (const float*)d_in[0];
    const int*   ei   = (const int*)  d_in[1];
    const int*   batch= (const int*)  d_in[2];
    const float* W1   = (const float*)d_in[3];
    const float* as1  = (const float*)d_in[4];
    const float* ad1  = (const float*)d_in[5];
    const float* g1   = (const float*)d_in[7];
    const float* be1  = (const float*)d_in[8];
    const float* W2   = (const float*)d_in[9];
    const float* as2  = (const float*)d_in[10];
    const float* ad2  = (const float*)d_in[11];
    const float* g2   = (const float*)d_in[13];
    const float* be2  = (const float*)d_in[14];
    const float* W3   = (const float*)d_in[15];
    const float* as3  = (const float*)d_in[16];
    const float* ad3  = (const float*)d_in[17];
    const float* g3   = (const float*)d_in[19];
    const float* be3  = (const float*)d_in[20];
    const float* Wc1  = (const float*)d_in[21];
    const float* bc1  = (const float*)d_in[22];
    const float* Wc2  = (const float*)d_in[23];
    const float* bc2  = (const float*)d_in[24];
    const float* We   = (const float*)d_in[25];
    const float* bee  = (const float*)d_in[26];

    const int n    = in_sizes[0] / 12;      // 100000
    const int Eraw = in_sizes[1] / 2;       // 1280000
    const int ET   = Eraw + n;              // + self loops
    const int Gn   = (out_size - n * HID) / 66;  // 64

    // workspace carve-out (256B aligned slots)
    char* ws = (char*)d_ws;
    size_t off = 0;
    auto carve = [&](size_t bytes) -> float* {
        float* p = (float*)(ws + off);
        off = (off + bytes + 255) & ~(size_t)255;
        return p;
    };
    float* feat = carve((size_t)n * HID * 4);
    float* hbuf = carve((size_t)n * HID * 4);
    float* agg  = carve((size_t)n * HID * 4);
    float* als  = carve((size_t)n * 4 * 4);
    float* ald  = carve((size_t)n * 4 * 4);
    float* mbuf = carve((size_t)n * 4 * 4);
    float* sbuf = carve((size_t)n * 4 * 4);
    float* eex  = carve((size_t)ET * 4 * 4);
    float* gsum = carve(HID * 4);
    float* gsq  = carve(HID * 4);
    float* psum = carve((size_t)Gn * HID * 4);
    float* pcnt = carve((size_t)Gn * 4);

    float* out_logits = (float*)d_out;
    float* out_emb    = (float*)d_out + (size_t)Gn * 2;
    float* nodeEmb    = (float*)d_out + (size_t)Gn * 2 + (size_t)Gn * HID;

    const float NEG_INF = -__builtin_inff();
    const int TB = 256;
    auto cdiv = [](int a, int b) { return (a + b - 1) / b; };

    struct LayerDesc {
        const float* Xin; int K; const float* W;
        const float* a_s; const float* a_d;
        const float* gm;  const float* bt;
        int heads; float* Yout;
    };
    LayerDesc layers[3] = {
        { x,    12, W1, as1, ad1, g1, be1, 4, feat    },
        { feat, 64, W2, as2, ad2, g2, be2, 4, feat    },
        { feat, 64, W3, as3, ad3, g3, be3, 1, nodeEmb },
    };

    for (int L = 0; L < 3; ++L) {
        const LayerDesc& ly = layers[L];
        const int heads = ly.heads;
        const int C = HID / heads;

        // 1. H = X @ W  (WMMA fp32 path)
        gemm_wmma<<<cdiv(n, 16), 128, 0, stream>>>(ly.Xin, ly.W, hbuf, n, ly.K);

        // 2. per-node attention coefficients
        attn_coef<<<cdiv(n * heads, TB), TB, 0, stream>>>(hbuf, ly.a_s, ly.a_d,
                                                          als, ald, n, heads, C);

        // 3. init segment-max / segment-sum / aggregation buffers
        fill_f32<<<cdiv(n * heads, TB), TB, 0, stream>>>(mbuf, NEG_INF, n * heads);
        fill_f32<<<cdiv(n * heads, TB), TB, 0, stream>>>(sbuf, 0.0f,    n * heads);
        fill_f32<<<cdiv(n * HID,   TB), TB, 0, stream>>>(agg,  0.0f,    n * HID);

        // 4-6. edge sweeps: score+max, exp+sum, weighted aggregation
        const int etot = ET * heads;
        edge_scores<<<cdiv(etot, TB), TB, 0, stream>>>(ei, als, ald, mbuf, eex,
                                                       ET, Eraw, heads);
        edge_expsum<<<cdiv(etot, TB), TB, 0, stream>>>(ei, eex, mbuf, sbuf,
                                                       ET, Eraw, heads);
        edge_aggr<<<cdiv(etot, TB), TB, 0, stream>>>(ei, hbuf, eex, sbuf, agg,
                                                     ET, Eraw, heads, C);

        // 7-9. BatchNorm + ELU (bias cancels inside BN)
        fill_f32<<<1, HID, 0, stream>>>(gsum, 0.0f, HID);
        fill_f32<<<1, HID, 0, stream>>>(gsq,  0.0f, HID);
        bn_stats<<<512, TB, 0, stream>>>(agg, gsum, gsq, n);
        bn_elu<<<cdiv(n * HID, TB), TB, 0, stream>>>(agg, gsum, gsq, ly.gm, ly.bt,
                                                     ly.Yout, n);
    }

    // global mean pool
    fill_f32<<<cdiv(Gn * HID, TB), TB, 0, stream>>>(psum, 0.0f, Gn * HID);
    fill_f32<<<cdiv(Gn, TB),       TB, 0, stream>>>(pcnt, 0.0f, Gn);
    pool_kernel<<<cdiv(n * HID, TB), TB, 0, stream>>>(nodeEmb, batch, psum, pcnt, n);

    // classifier + embedding heads
    head_kernel<<<cdiv(Gn, 64), 64, 0, stream>>>(psum, pcnt, Wc1, bc1, Wc2, bc2,
                                                 We, bee, out_logits, out_emb, Gn);
}